// AgentAttention_42013370089893
// MI455X (gfx1250) — compile-verified
//
#include <hip/hip_runtime.h>
#include <hip/hip_bf16.h>
#include <math.h>

#define B_   32
#define H0_  56
#define W0_  56
#define N_   3136
#define C_   256
#define NH_  8
#define HD_  32
#define A_   49
#define P_   7
#define C3_  768
#define SCALE 0.17677669529663687f   // 32^-0.5

typedef __attribute__((ext_vector_type(2))) float v2f;
typedef __attribute__((ext_vector_type(8))) float v8f;

__device__ __forceinline__ v8f wmma_f32(v2f a, v2f b, v8f c) {
  // V_WMMA_F32_16X16X4_F32 : D = A(16x4) * B(4x16) + C
  return __builtin_amdgcn_wmma_f32_16x16x4_f32(false, a, false, b, (short)0, c,
                                               false, false);
}

// ------------------------------------------- CDNA5 async global->LDS staging
#if __has_builtin(__builtin_amdgcn_global_load_async_to_lds_b128)
#define HAVE_ASYNC 1
#else
#define HAVE_ASYNC 0
#endif

__device__ __forceinline__ void async_copy16(const float* g, float* l) {
#if HAVE_ASYNC
  typedef int v4i __attribute__((vector_size(16)));
  __builtin_amdgcn_global_load_async_to_lds_b128(
      (__attribute__((address_space(1))) v4i*)g,
      (__attribute__((address_space(3))) v4i*)l, 0, 0);
#else
  *(float4*)l = *(const float4*)g;
#endif
}

__device__ __forceinline__ void wait_async() {
#if HAVE_ASYNC
#if __has_builtin(__builtin_amdgcn_s_wait_asynccnt)
  __builtin_amdgcn_s_wait_asynccnt(0);
#else
  asm volatile("s_wait_asynccnt 0" ::: "memory");
#endif
#endif
}

// ---------------------------------------------------------------- bias prep
__device__ __forceinline__ float bilerp7(const float* __restrict__ src, int y, int x) {
  float sy = (y + 0.5f) * 0.125f - 0.5f;
  float sx = (x + 0.5f) * 0.125f - 0.5f;
  float fy = floorf(sy), fx = floorf(sx);
  float wy = sy - fy,   wx = sx - fx;
  int y0 = (int)fy, x0 = (int)fx;
  int y0c = min(max(y0, 0), 6), y1c = min(max(y0 + 1, 0), 6);
  int x0c = min(max(x0, 0), 6), x1c = min(max(x0 + 1, 0), 6);
  float v00 = src[y0c * 7 + x0c], v01 = src[y0c * 7 + x1c];
  float v10 = src[y1c * 7 + x0c], v11 = src[y1c * 7 + x1c];
  return v00 * (1.f - wy) * (1.f - wx) + v01 * (1.f - wy) * wx +
         v10 * wy * (1.f - wx) + v11 * wy * wx;
}

__global__ __launch_bounds__(256)
void bias_kernel(const float* __restrict__ an_bias, const float* __restrict__ na_bias,
                 const float* __restrict__ ah_bias, const float* __restrict__ aw_bias,
                 const float* __restrict__ ha_bias, const float* __restrict__ wa_bias,
                 float* __restrict__ pos_bias, float* __restrict__ agent_bias) {
  int idx = blockIdx.x * blockDim.x + threadIdx.x;
  if (idx >= NH_ * A_ * N_) return;
  int t  = idx % N_;
  int ha = idx / N_;            // h*49 + a
  int a  = ha % A_;
  int h  = ha / A_;
  int y = t / W0_, x = t % W0_;
  float pb = bilerp7(an_bias + (size_t)ha * 49, y, x) +
             ah_bias[(size_t)ha * H0_ + y] + aw_bias[(size_t)ha * W0_ + x];
  pos_bias[(size_t)ha * N_ + t] = pb;                       // (NH, A, N)
  float ab = bilerp7(na_bias + (size_t)ha * 49, y, x) +
             ha_bias[((size_t)h * H0_ + y) * A_ + a] +
             wa_bias[((size_t)h * W0_ + x) * A_ + a];
  agent_bias[((size_t)h * N_ + t) * A_ + a] = ab;           // (NH, N, A)
}

// ---------------------------------------------- generic GEMM: D = A*B + bias
// M mult of 128, K mult of 32, Ncol mult of 64. Block = 256 thr = 8 waves.
// Block tile 128x64: wave w owns rows [w*16,w*16+16) x all 64 cols
// (4 accumulators). B K-chunks staged in LDS via async global->LDS copies.
#define GEMM_KC 32
__global__ __launch_bounds__(256)
void gemm_bias_kernel(const float* __restrict__ Am, const float* __restrict__ Bm,
                      const float* __restrict__ bias, float* __restrict__ Dm,
                      int M, int K, int Ncol) {
  __shared__ __align__(16) float LB[GEMM_KC][68];   // +4 pad: half-wave bank split
  int tid = threadIdx.x;
  int lane = tid & 31, wave = tid >> 5;
  int colBase = blockIdx.x * 64;
  int rowBase = blockIdx.y * 128 + wave * 16;
  int m = lane & 15, kb = (lane >> 4) * 2;
  const float* arow = Am + (size_t)(rowBase + m) * K;
  v8f acc0 = {}, acc1 = {}, acc2 = {}, acc3 = {};
  for (int kc = 0; kc < K; kc += GEMM_KC) {
    #pragma unroll
    for (int j = 0; j < 2; ++j) {          // 512 x b128 units, 2 per thread
      int i = tid * 2 + j;
      int r = i >> 4, cc = (i & 15) * 4;
      async_copy16(Bm + (size_t)(kc + r) * Ncol + colBase + cc, &LB[r][cc]);
    }
    wait_async();
    __syncthreads();
    #pragma unroll
    for (int kk = 0; kk < GEMM_KC; kk += 4) {
      v2f a;
      a.x = arow[kc + kk + kb];
      a.y = arow[kc + kk + kb + 1];
      const float* lb0 = &LB[kk + kb][m];
      const float* lb1 = &LB[kk + kb + 1][m];
      v2f b0, b1, b2, b3;
      b0.x = lb0[0];  b0.y = lb1[0];
      b1.x = lb0[16]; b1.y = lb1[16];
      b2.x = lb0[32]; b2.y = lb1[32];
      b3.x = lb0[48]; b3.y = lb1[48];
      acc0 = wmma_f32(a, b0, acc0);
      acc1 = wmma_f32(a, b1, acc1);
      acc2 = wmma_f32(a, b2, acc2);
      acc3 = wmma_f32(a, b3, acc3);
    }
    __syncthreads();
  }
  int r0 = rowBase + 8 * (lane >> 4);
  #pragma unroll
  for (int nt = 0; nt < 4; ++nt) {
    int col = colBase + nt * 16 + m;
    float bv = bias[col];
    v8f acc = nt == 0 ? acc0 : nt == 1 ? acc1 : nt == 2 ? acc2 : acc3;
    #pragma unroll
    for (int r = 0; r < 8; ++r)
      Dm[(size_t)(r0 + r) * Ncol + col] = acc[r] + bv;
  }
}

// ------------------------------------------------------------- agent pooling
__global__ __launch_bounds__(256)
void agent_pool_kernel(const float* __restrict__ qkv, float* __restrict__ agent) {
  int idx = blockIdx.x * blockDim.x + threadIdx.x;
  if (idx >= B_ * A_ * C_) return;
  int c = idx % C_;
  int a = (idx / C_) % A_;
  int b = idx / (C_ * A_);
  int py = a / P_, px = a % P_;
  float s = 0.f;
  #pragma unroll 4
  for (int iy = 0; iy < 8; ++iy)
    for (int ix = 0; ix < 8; ++ix) {
      int t = (py * 8 + iy) * W0_ + px * 8 + ix;
      s += qkv[((size_t)b * N_ + t) * C3_ + c];   // q slice: cols [0,256)
    }
  agent[idx] = s * (1.0f / 64.0f);                // (B, A, C)
}

// --------------------------------- stage 1: S = scale*Agent @ K^T + pos_bias
__global__ __launch_bounds__(256)
void s1_scores_kernel(const float* __restrict__ qkv, const float* __restrict__ agent,
                      const float* __restrict__ pos_bias, float* __restrict__ scores) {
  int bh = blockIdx.x;
  int b = bh >> 3, h = bh & 7;
  int lane = threadIdx.x & 31, wave = threadIdx.x >> 5;
  int tileId = blockIdx.y * 8 + wave;
  int Mtile = tileId & 3;
  int Ntile = tileId >> 2;
  int m = lane & 15, kb = (lane >> 4) * 2;
  int arow = Mtile * 16 + m;
  int tcol = Ntile * 16 + m;
  bool av = arow < A_;
  const float* aptr = agent + ((size_t)(b * A_ + (av ? arow : 0))) * C_ + h * HD_ + kb;
  const float* kptr = qkv + ((size_t)b * N_ + tcol) * C3_ + C_ + h * HD_ + kb;
  v8f acc = {};
  #pragma unroll
  for (int kk = 0; kk < HD_; kk += 4) {
    v2f a, bf;
    a.x = av ? aptr[kk] * SCALE : 0.f;
    a.y = av ? aptr[kk + 1] * SCALE : 0.f;
    bf.x = kptr[kk];
    bf.y = kptr[kk + 1];
    acc = wmma_f32(a, bf, acc);
  }
  int col = Ntile * 16 + m;
  int r0 = Mtile * 16 + 8 * (lane >> 4);
  #pragma unroll
  for (int r = 0; r < 8; ++r) {
    int row = r0 + r;
    if (row < A_)
      scores[((size_t)bh * A_ + row) * N_ + col] =
          acc[r] + pos_bias[((size_t)h * A_ + row) * N_ + col];
  }
}

// ------------------------------------------------------- row softmax (len N)
__global__ __launch_bounds__(256)
void softmax_rows_kernel(float* __restrict__ scores) {
  __shared__ float red[256];
  float* p = scores + (size_t)blockIdx.x * N_;
  int tid = threadIdx.x;
  float mx = -3.0e38f;
  for (int i = tid; i < N_; i += 256) mx = fmaxf(mx, p[i]);
  red[tid] = mx; __syncthreads();
  for (int s = 128; s > 0; s >>= 1) { if (tid < s) red[tid] = fmaxf(red[tid], red[tid + s]); __syncthreads(); }
  mx = red[0]; __syncthreads();
  float sum = 0.f;
  for (int i = tid; i < N_; i += 256) { float e = __expf(p[i] - mx); p[i] = e; sum += e; }
  red[tid] = sum; __syncthreads();
  for (int s = 128; s > 0; s >>= 1) { if (tid < s) red[tid] += red[tid + s]; __syncthreads(); }
  float inv = 1.0f / red[0];
  for (int i = tid; i < N_; i += 256) p[i] *= inv;
}

// ----------------------------------- stage 1b: agent_v = attn(49xN) @ V(Nx32)
__global__ __launch_bounds__(256)
void s1_av_kernel(const float* __restrict__ scores, const float* __restrict__ qkv,
                  float* __restrict__ agent_v) {
  int bh = blockIdx.x;
  int b = bh >> 3, h = bh & 7;
  int lane = threadIdx.x & 31, wave = threadIdx.x >> 5;
  int Mtile = wave >> 1, Ntile = wave & 1;
  int m = lane & 15, kb = (lane >> 4) * 2;
  int arow = Mtile * 16 + m;
  bool av = arow < A_;
  const float* aptr = scores + ((size_t)bh * A_ + (av ? arow : 0)) * N_ + kb;
  const float* vptr = qkv + ((size_t)b * N_ + kb) * C3_ + 2 * C_ + h * HD_ + Ntile * 16 + m;
  v8f acc = {};
  #pragma unroll 8
  for (int kk = 0; kk < N_; kk += 4) {
    v2f a, bf;
    a.x = av ? aptr[kk] : 0.f;
    a.y = av ? aptr[kk + 1] : 0.f;
    bf.x = vptr[(size_t)kk * C3_];
    bf.y = vptr[(size_t)(kk + 1) * C3_];
    acc = wmma_f32(a, bf, acc);
  }
  int r0 = Mtile * 16 + 8 * (lane >> 4);
  #pragma unroll
  for (int r = 0; r < 8; ++r) {
    int row = r0 + r;
    if (row < A_)
      agent_v[((size_t)bh * A_ + row) * HD_ + Ntile * 16 + m] = acc[r];
  }
}

// ---- stage 2 fused: S = scale*Q@Agent^T + bias ; softmax(49) ; O = P@agent_v
__global__ __launch_bounds__(256)
void s2_kernel(const float* __restrict__ qkv, const float* __restrict__ agent,
               const float* __restrict__ agent_bias, const float* __restrict__ agent_v,
               float* __restrict__ out) {
  __shared__ float S[64][65];
  int bh = blockIdx.x;
  int b = bh >> 3, h = bh & 7;
  int tBase = blockIdx.y * 64;
  int lane = threadIdx.x & 31, wave = threadIdx.x >> 5;
  int m = lane & 15, kb = (lane >> 4) * 2;

  { // phase 1: S(64x64) = scale*Q(64x32) @ Agent^T(32x64) + agent_bias
    int Mtile = wave >> 1;
    int trow = tBase + Mtile * 16 + m;
    const float* qptr = qkv + ((size_t)b * N_ + trow) * C3_ + h * HD_ + kb;
    #pragma unroll
    for (int nt = 0; nt < 2; ++nt) {
      int Ntile = (wave & 1) * 2 + nt;
      int acol = Ntile * 16 + m;
      bool gv = acol < A_;
      const float* gptr = agent + ((size_t)(b * A_ + (gv ? acol : 0))) * C_ + h * HD_ + kb;
      v8f acc = {};
      #pragma unroll
      for (int kk = 0; kk < HD_; kk += 4) {
        v2f a, bf;
        a.x = qptr[kk] * SCALE;  a.y = qptr[kk + 1] * SCALE;
        bf.x = gv ? gptr[kk] : 0.f;
        bf.y = gv ? gptr[kk + 1] : 0.f;
        acc = wmma_f32(a, bf, acc);
      }
      int c  = Ntile * 16 + m;
      int r0 = Mtile * 16 + 8 * (lane >> 4);
      #pragma unroll
      for (int r = 0; r < 8; ++r) {
        int rw = r0 + r;
        float val = -1.0e30f;
        if (c < A_)
          val = acc[r] + agent_bias[((size_t)h * N_ + tBase + rw) * A_ + c];
        S[rw][c] = val;
      }
    }
  }
  __syncthreads();

  if (threadIdx.x < 64) { // phase 2: softmax over 49 agents (padding -> 0)
    int rw = threadIdx.x;
    float mx = -1.0e30f;
    #pragma unroll 8
    for (int j = 0; j < 64; ++j) mx = fmaxf(mx, S[rw][j]);
    float sum = 0.f;
    #pragma unroll 8
    for (int j = 0; j < 64; ++j) { float e = __expf(S[rw][j] - mx); S[rw][j] = e; sum += e; }
    float inv = 1.0f / sum;
    #pragma unroll 8
    for (int j = 0; j < 64; ++j) S[rw][j] *= inv;
  }
  __syncthreads();

  { // phase 3: O(64x32) = P(64x64) @ agent_v(64x32)
    int Mtile = wave & 3, Ntile = wave >> 2;
    v8f acc = {};
    #pragma unroll
    for (int kk = 0; kk < 64; kk += 4) {
      int k0 = kk + kb;
      v2f a, bf;
      a.x = S[Mtile * 16 + m][k0];
      a.y = S[Mtile * 16 + m][k0 + 1];
      bf.x = (k0     < A_) ? agent_v[((size_t)bh * A_ + k0    ) * HD_ + Ntile * 16 + m] : 0.f;
      bf.y = (k0 + 1 < A_) ? agent_v[((size_t)bh * A_ + k0 + 1) * HD_ + Ntile * 16 + m] : 0.f;
      acc = wmma_f32(a, bf, acc);
    }
    int r0 = Mtile * 16 + 8 * (lane >> 4);
    int d  = Ntile * 16 + m;
    #pragma unroll
    for (int r = 0; r < 8; ++r) {
      int trow = tBase + r0 + r;
      out[((size_t)b * N_ + trow) * C_ + h * HD_ + d] = acc[r];
    }
  }
}

// ---------------------------------------------- depthwise 3x3 conv (+= into out)
__global__ __launch_bounds__(256)
void dwc_kernel(const float* __restrict__ qkv, const float* __restrict__ dwc_w,
                const float* __restrict__ dwc_b, float* __restrict__ out) {
  size_t idx = (size_t)blockIdx.x * blockDim.x + threadIdx.x;
  if (idx >= (size_t)B_ * N_ * C_) return;
  int c = (int)(idx % C_);
  int t = (int)((idx / C_) % N_);
  int b = (int)(idx / ((size_t)C_ * N_));
  int y = t / W0_, x = t % W0_;
  float s = dwc_b[c];
  #pragma unroll
  for (int ky = 0; ky < 3; ++ky) {
    int yy = y + ky - 1;
    if (yy < 0 || yy >= H0_) continue;
    #pragma unroll
    for (int kx = 0; kx < 3; ++kx) {
      int xx = x + kx - 1;
      if (xx < 0 || xx >= W0_) continue;
      s += qkv[((size_t)b * N_ + yy * W0_ + xx) * C3_ + 2 * C_ + c] *
           dwc_w[c * 9 + ky * 3 + kx];
    }
  }
  out[idx] += s;
}

// ---------------------------------------------------------------------------
extern "C" void kernel_launch(void* const* d_in, const int* in_sizes, int n_in,
                              void* d_out, int out_size, void* d_ws, size_t ws_size,
                              hipStream_t stream) {
  const float* x       = (const float*)d_in[0];
  const float* qkv_w   = (const float*)d_in[1];
  const float* qkv_b   = (const float*)d_in[2];
  const float* proj_w  = (const float*)d_in[3];
  const float* proj_b  = (const float*)d_in[4];
  const float* dwc_w   = (const float*)d_in[5];
  const float* dwc_b   = (const float*)d_in[6];
  const float* an_bias = (const float*)d_in[7];
  const float* na_bias = (const float*)d_in[8];
  const float* ah_bias = (const float*)d_in[9];
  const float* aw_bias = (const float*)d_in[10];
  const float* ha_bias = (const float*)d_in[11];
  const float* wa_bias = (const float*)d_in[12];
  float* outp = (float*)d_out;

  float* ws = (float*)d_ws;
  size_t o = 0;
  float* qkv        = ws + o;  o += (size_t)B_ * N_ * C3_;       // 77,070,336
  float* pos_bias   = ws + o;  o += (size_t)NH_ * A_ * N_;       //  1,229,312
  float* agent_bias = ws + o;  o += (size_t)NH_ * N_ * A_;       //  1,229,312
  float* agent      = ws + o;  o += (size_t)B_ * A_ * C_;        //    401,408
  float* scores     = ws + o;  o += (size_t)B_ * NH_ * A_ * N_;  // 39,337,984
  float* agent_v    = ws + o;  o += (size_t)B_ * NH_ * A_ * HD_; //    401,408
  float* attn       = ws + o;  o += (size_t)B_ * N_ * C_;        // 25,690,112

  // 1) position / agent biases (resize math collapsed analytically)
  {
    int n = NH_ * A_ * N_;
    bias_kernel<<<(n + 255) / 256, 256, 0, stream>>>(
        an_bias, na_bias, ah_bias, aw_bias, ha_bias, wa_bias, pos_bias, agent_bias);
  }
  // 2) qkv = x @ qkv_w + qkv_b    (100352 x 768 x 256)
  gemm_bias_kernel<<<dim3(C3_ / 64, (B_ * N_) / 128), 256, 0, stream>>>(
      x, qkv_w, qkv_b, qkv, B_ * N_, C_, C3_);
  // 3) agent tokens = 8x8 avg-pool of q
  agent_pool_kernel<<<(B_ * A_ * C_ + 255) / 256, 256, 0, stream>>>(qkv, agent);
  // 4) stage-1 scores, softmax, agent_v
  s1_scores_kernel<<<dim3(B_ * NH_, 98), 256, 0, stream>>>(qkv, agent, pos_bias, scores);
  softmax_rows_kernel<<<B_ * NH_ * A_, 256, 0, stream>>>(scores);
  s1_av_kernel<<<B_ * NH_, 256, 0, stream>>>(scores, qkv, agent_v);
  // 5) stage-2 fused attention
  s2_kernel<<<dim3(B_ * NH_, N_ / 64), 256, 0, stream>>>(qkv, agent, agent_bias, agent_v, attn);
  // 6) depthwise conv accumulate
  dwc_kernel<<<(unsigned)(((size_t)B_ * N_ * C_ + 255) / 256), 256, 0, stream>>>(
      qkv, dwc_w, dwc_b, attn);
  // 7) out = attn @ proj_w + proj_b
  gemm_bias_kernel<<<dim3(C_ / 64, (B_ * N_) / 128), 256, 0, stream>>>(
      attn, proj_w, proj_b, outp, B_ * N_, C_, C_);
}